// GATLayer_63204738728334
// MI455X (gfx1250) — compile-verified
//
#include <hip/hip_runtime.h>
#include <math.h>

typedef float v2f __attribute__((ext_vector_type(2)));
typedef float v8f __attribute__((ext_vector_type(8)));

#define N_NODES   50000
#define N_EDGES   800000
#define N_ETOT    (N_EDGES + N_NODES)
#define CH        128          // IN_CH == H*C == 128
#define HEADS     4
#define NEG_SLOPE 0.2f
#define LN_EPS    1e-5f

// ---------- ordered-uint encoding for float atomicMax ----------
__device__ __forceinline__ unsigned int fkey(float f) {
    unsigned int u = __float_as_uint(f);
    return (u & 0x80000000u) ? ~u : (u | 0x80000000u);
}
__device__ __forceinline__ float fdec(unsigned int k) {
    unsigned int u = (k & 0x80000000u) ? (k & 0x7FFFFFFFu) : ~k;
    return __uint_as_float(u);
}
#define FKEY_NEG_INF 0x007FFFFFu   // fkey(-inf)

// ---------- 0: init workspace ----------
__global__ void init_ws_kernel(float* __restrict__ agg,
                               float* __restrict__ segsum,
                               unsigned int* __restrict__ segmax) {
    int i = blockIdx.x * blockDim.x + threadIdx.x;
    if (i < N_NODES * CH) agg[i] = 0.0f;
    if (i < N_NODES * HEADS) {
        segsum[i] = 0.0f;
        segmax[i] = FKEY_NEG_INF;
    }
}

// ---------- 1: fused dual projection GEMM via V_WMMA_F32_16X16X4_F32 ----------
// One wave (32 lanes) per (16-row tile, {L|R}); computes all 8 column tiles.
__global__ __launch_bounds__(32) void gemm_proj_kernel(
    const float* __restrict__ x,
    const float* __restrict__ w_l, const float* __restrict__ b_l,
    const float* __restrict__ w_r, const float* __restrict__ b_r,
    float* __restrict__ xl, float* __restrict__ xr) {
    const int bid = blockIdx.x;
    const int mat = bid & 1;        // 0 -> L, 1 -> R
    const int mt  = bid >> 1;       // row tile index [0, 3125)
    const float* __restrict__ W = mat ? w_r : w_l;
    const float* __restrict__ B = mat ? b_r : b_l;
    float* __restrict__ Y       = mat ? xr  : xl;

    const int lane = threadIdx.x & 31;
    const int rc   = lane & 15;            // A: row M, B: col N
    const int kk   = (lane >> 4) << 1;     // 0 or 2 (K sub-pair)

    const float* __restrict__ xrow = x + (size_t)(mt * 16 + rc) * CH;

    v8f acc[8];
#pragma unroll
    for (int n = 0; n < 8; ++n) acc[n] = (v8f){0, 0, 0, 0, 0, 0, 0, 0};

    for (int k0 = 0; k0 < CH; k0 += 4) {
        // A fragment: X[mt*16 + rc][k0+kk .. k0+kk+1]  (8B aligned)
        v2f a = *(const v2f*)(xrow + k0 + kk);
#pragma unroll
        for (int n = 0; n < 8; ++n) {
            v2f b;
            b.x = W[(size_t)(k0 + kk)     * CH + n * 16 + rc];
            b.y = W[(size_t)(k0 + kk + 1) * CH + n * 16 + rc];
            acc[n] = __builtin_amdgcn_wmma_f32_16x16x4_f32(
                false, a, false, b, (short)0, acc[n], false, false);
        }
    }

    // C/D layout: VGPR v, lane L -> M = v + 8*(L/16), N = L%16
    const int mrow = mt * 16 + 8 * (lane >> 4);
#pragma unroll
    for (int n = 0; n < 8; ++n) {
        const float bias = B[n * 16 + rc];
#pragma unroll
        for (int v = 0; v < 8; ++v) {
            Y[(size_t)(mrow + v) * CH + n * 16 + rc] = acc[n][v] + bias;
        }
    }
}

// ---------- 2: per-edge GATv2 logits + segment max ----------
// One wave per edge; lane covers 4 features; lanes [8h, 8h+8) -> head h.
__global__ void edge_logits_kernel(const float* __restrict__ xl,
                                   const float* __restrict__ xr,
                                   const int* __restrict__ ei,
                                   const float* __restrict__ att,
                                   float* __restrict__ logits,
                                   unsigned int* __restrict__ segmax) {
    int e = blockIdx.x * (blockDim.x >> 5) + (threadIdx.x >> 5);
    if (e >= N_ETOT) return;
    const int lane = threadIdx.x & 31;

    int src, dst;
    if (e < N_EDGES) { src = ei[e]; dst = ei[N_EDGES + e]; }
    else             { src = dst = e - N_EDGES; }

    float4 l = ((const float4*)(xl + (size_t)src * CH))[lane];
    float4 r = ((const float4*)(xr + (size_t)dst * CH))[lane];
    float4 a = ((const float4*)att)[lane];

    float t, s = 0.0f;
    t = l.x + r.x; t = t > 0.0f ? t : t * NEG_SLOPE; s += t * a.x;
    t = l.y + r.y; t = t > 0.0f ? t : t * NEG_SLOPE; s += t * a.y;
    t = l.z + r.z; t = t > 0.0f ? t : t * NEG_SLOPE; s += t * a.z;
    t = l.w + r.w; t = t > 0.0f ? t : t * NEG_SLOPE; s += t * a.w;

    // reduce within each 8-lane head group
    s += __shfl_xor(s, 1, 8);
    s += __shfl_xor(s, 2, 8);
    s += __shfl_xor(s, 4, 8);

    if ((lane & 7) == 0) {
        const int h = lane >> 3;
        logits[(size_t)e * HEADS + h] = s;
        atomicMax(&segmax[(size_t)dst * HEADS + h], fkey(s));
    }
}

// ---------- 3: exp(logit - max) + segment sum ----------
__global__ void edge_exp_kernel(const int* __restrict__ ei,
                                float* __restrict__ logits,      // in: logit, out: expv
                                const unsigned int* __restrict__ segmax,
                                float* __restrict__ segsum) {
    int idx = blockIdx.x * blockDim.x + threadIdx.x;
    if (idx >= N_ETOT * HEADS) return;
    const int e = idx >> 2;
    const int h = idx & 3;
    const int dst = (e < N_EDGES) ? ei[N_EDGES + e] : (e - N_EDGES);
    const float m = fdec(segmax[(size_t)dst * HEADS + h]);
    const float v = expf(logits[idx] - m);
    logits[idx] = v;
    atomicAdd(&segsum[(size_t)dst * HEADS + h], v);
}

// ---------- 4: alpha * xl[src] scatter-add ----------
__global__ void edge_aggregate_kernel(const float* __restrict__ xl,
                                      const int* __restrict__ ei,
                                      const float* __restrict__ expv,
                                      const float* __restrict__ segsum,
                                      float* __restrict__ agg) {
    int e = blockIdx.x * (blockDim.x >> 5) + (threadIdx.x >> 5);
    if (e >= N_ETOT) return;
    const int lane = threadIdx.x & 31;

    int src, dst;
    if (e < N_EDGES) { src = ei[e]; dst = ei[N_EDGES + e]; }
    else             { src = dst = e - N_EDGES; }

    const int h = lane >> 3;
    const float alpha = expv[(size_t)e * HEADS + h] /
                        segsum[(size_t)dst * HEADS + h];

    float4 l = ((const float4*)(xl + (size_t)src * CH))[lane];
    float* o = agg + (size_t)dst * CH + lane * 4;
    atomicAdd(o + 0, alpha * l.x);
    atomicAdd(o + 1, alpha * l.y);
    atomicAdd(o + 2, alpha * l.z);
    atomicAdd(o + 3, alpha * l.w);
}

// ---------- 5: bias + LayerNorm, one wave per node ----------
__global__ void layernorm_kernel(const float* __restrict__ agg,
                                 const float* __restrict__ bias,
                                 const float* __restrict__ gamma,
                                 const float* __restrict__ beta,
                                 float* __restrict__ out) {
    int n = blockIdx.x * (blockDim.x >> 5) + (threadIdx.x >> 5);
    if (n >= N_NODES) return;
    const int lane = threadIdx.x & 31;

    float4 v = ((const float4*)(agg + (size_t)n * CH))[lane];
    float4 b = ((const float4*)bias)[lane];
    v.x += b.x; v.y += b.y; v.z += b.z; v.w += b.w;

    float sum = v.x + v.y + v.z + v.w;
    float sq  = v.x * v.x + v.y * v.y + v.z * v.z + v.w * v.w;
#pragma unroll
    for (int off = 1; off < 32; off <<= 1) {
        sum += __shfl_xor(sum, off, 32);
        sq  += __shfl_xor(sq,  off, 32);
    }
    const float mu   = sum * (1.0f / CH);
    const float var  = sq * (1.0f / CH) - mu * mu;
    const float rstd = rsqrtf(var + LN_EPS);

    float4 g  = ((const float4*)gamma)[lane];
    float4 bt = ((const float4*)beta)[lane];
    float4 o;
    o.x = (v.x - mu) * rstd * g.x + bt.x;
    o.y = (v.y - mu) * rstd * g.y + bt.y;
    o.z = (v.z - mu) * rstd * g.z + bt.z;
    o.w = (v.w - mu) * rstd * g.w + bt.w;
    ((float4*)(out + (size_t)n * CH))[lane] = o;
}

extern "C" void kernel_launch(void* const* d_in, const int* in_sizes, int n_in,
                              void* d_out, int out_size, void* d_ws, size_t ws_size,
                              hipStream_t stream) {
    const float* x      = (const float*)d_in[0];
    const int*   ei     = (const int*)d_in[1];
    const float* w_l    = (const float*)d_in[2];
    const float* b_l    = (const float*)d_in[3];
    const float* w_r    = (const float*)d_in[4];
    const float* b_r    = (const float*)d_in[5];
    const float* att    = (const float*)d_in[6];
    const float* bias   = (const float*)d_in[7];
    const float* gamma  = (const float*)d_in[8];
    const float* beta   = (const float*)d_in[9];
    float* out = (float*)d_out;

    // workspace carve-up (all 256B aligned)
    char* base = (char*)d_ws;
    size_t off = 0;
    float* xl = (float*)(base + off);           off += (size_t)N_NODES * CH * 4;   // 25.6 MB
    float* xr = (float*)(base + off);           off += (size_t)N_NODES * CH * 4;   // 25.6 MB
    float* logits = (float*)(base + off);       off += (size_t)N_ETOT * HEADS * 4; // 13.6 MB
    unsigned int* segmax = (unsigned int*)(base + off); off += (size_t)N_NODES * HEADS * 4;
    float* segsum = (float*)(base + off);       off += (size_t)N_NODES * HEADS * 4;
    float* agg = (float*)(base + off);          off += (size_t)N_NODES * CH * 4;   // 25.6 MB
    (void)ws_size; (void)in_sizes; (void)n_in; (void)out_size;

    // 0: init
    {
        int threads = 256;
        int total = N_NODES * CH;
        init_ws_kernel<<<(total + threads - 1) / threads, threads, 0, stream>>>(
            agg, segsum, segmax);
    }
    // 1: projections (WMMA)
    {
        int mtiles = N_NODES / 16;  // 3125 exact
        gemm_proj_kernel<<<mtiles * 2, 32, 0, stream>>>(x, w_l, b_l, w_r, b_r, xl, xr);
    }
    // 2: logits + segment max
    {
        int waves_per_block = 8;                  // 256 threads
        int blocks = (N_ETOT + waves_per_block - 1) / waves_per_block;
        edge_logits_kernel<<<blocks, waves_per_block * 32, 0, stream>>>(
            xl, xr, ei, att, logits, segmax);
    }
    // 3: exp + segment sum
    {
        int threads = 256;
        int total = N_ETOT * HEADS;
        edge_exp_kernel<<<(total + threads - 1) / threads, threads, 0, stream>>>(
            ei, logits, segmax, segsum);
    }
    // 4: scatter-add aggregation
    {
        int waves_per_block = 8;
        int blocks = (N_ETOT + waves_per_block - 1) / waves_per_block;
        edge_aggregate_kernel<<<blocks, waves_per_block * 32, 0, stream>>>(
            xl, ei, logits, segsum, agg);
    }
    // 5: bias + LayerNorm
    {
        int waves_per_block = 8;
        int blocks = (N_NODES + waves_per_block - 1) / waves_per_block;
        layernorm_kernel<<<blocks, waves_per_block * 32, 0, stream>>>(
            agg, bias, gamma, beta, out);
    }
}